// GCN_3667902071144
// MI455X (gfx1250) — compile-verified
//
#include <hip/hip_runtime.h>

typedef float v2f __attribute__((ext_vector_type(2)));
typedef float v8f __attribute__((ext_vector_type(8)));

#define N_NODES 100000
#define N_EDGES 1600000
#define F_IN    128
#define HID     64
#define C_OUT   16

// ---------------- init ----------------
__global__ void init_deg_kernel(float* deg, unsigned long long* mp) {
    int i = blockIdx.x * 256 + threadIdx.x;
    if (i < N_NODES) deg[i] = 1.0f;               // +1 for self loop
    if (i == 0) *mp = ~0ull;                      // argmin accumulator
}

__global__ void init_zero_kernel(float* p, int n) {
    int i = blockIdx.x * 256 + threadIdx.x;
    if (i < n) p[i] = 0.0f;
}

// ---------------- degree + argmin over row==0 edges ----------------
__global__ void edge_deg_kernel(const long long* __restrict__ ei,
                                const float* __restrict__ ew,
                                float* deg, unsigned long long* mp) {
    int e = blockIdx.x * 256 + threadIdx.x;
    if (e >= N_EDGES) return;
    int r = (int)ei[e];
    int c = (int)ei[N_EDGES + e];
    float w = ew[e];
    atomicAdd(&deg[c], w);
    if (r == 0) {
        // weights are uniform [0,1): positive, so uint order == float order.
        unsigned long long pk =
            ((unsigned long long)__float_as_uint(w) << 32) | (unsigned)e;
        atomicMin(mp, pk);                        // min weight, lowest index tiebreak
    }
}

__global__ void dinv_kernel(const float* __restrict__ deg, float* d1, float* d2) {
    int i = blockIdx.x * 256 + threadIdx.x;
    if (i < N_NODES) {
        float v = rsqrtf(deg[i]);
        d1[i] = v;
        d2[i] = v;
    }
}

// single thread: resolve deleted edge, patch layer-2 dinv at its col
__global__ void fixup_kernel(const long long* __restrict__ ei,
                             const float* __restrict__ ew,
                             const float* __restrict__ deg,
                             const unsigned long long* mp,
                             float* d2, int* del_idx) {
    unsigned long long pk = *mp;
    int idx = (pk == ~0ull) ? 0 : (int)(unsigned)(pk & 0xffffffffull); // argmin of all-inf -> 0
    *del_idx = idx;
    int c = (int)ei[N_EDGES + idx];
    float w = ew[idx];
    d2[c] = rsqrtf(deg[c] - w);                   // deg2 = deg1 - deleted weight at col
}

// ---------------- GEMM1: h0 = x @ W1  (fp32 WMMA 16x16x4) ----------------
__global__ void __launch_bounds__(256)
gemm1_wmma(const float* __restrict__ x, const float* __restrict__ W,
           float* __restrict__ h0) {
    int wave = threadIdx.x >> 5;                  // 8 waves: 2 row tiles x 4 col tiles
    int lane = threadIdx.x & 31;
    int m0 = (blockIdx.x * 2 + (wave >> 2)) * 16; // 6250 row tiles total (exact)
    int n0 = (wave & 3) * 16;
    int m  = lane & 15;
    int hh = lane >> 4;                           // K-half select per ISA A layout
    v8f c = {};
    const float* arow = x + (size_t)(m0 + m) * F_IN;
    for (int k0 = 0; k0 < F_IN; k0 += 4) {
        int k = k0 + 2 * hh;
        v2f a; a.x = arow[k];                 a.y = arow[k + 1];
        v2f b; b.x = W[(size_t)k * HID + n0 + m];
               b.y = W[(size_t)(k + 1) * HID + n0 + m];
        c = __builtin_amdgcn_wmma_f32_16x16x4_f32(false, a, false, b,
                                                  (short)0, c, false, false);
    }
    // D layout: VGPR i -> row m0 + i + 8*hh, col n0 + (lane&15)
    float* drow = h0 + (size_t)(m0 + 8 * hh) * HID + n0 + m;
#pragma unroll
    for (int i = 0; i < 8; ++i) drow[(size_t)i * HID] = c[i];
}

// ---------------- GEMM2: h2 = h @ W2  (fp32 WMMA 16x16x4) ----------------
__global__ void __launch_bounds__(256)
gemm2_wmma(const float* __restrict__ h, const float* __restrict__ W,
           float* __restrict__ h2) {
    int wave = threadIdx.x >> 5;
    int lane = threadIdx.x & 31;
    int m0 = (blockIdx.x * 8 + wave) * 16;
    if (m0 >= N_NODES) return;                    // wave-uniform guard (EXEC stays full)
    int m  = lane & 15;
    int hh = lane >> 4;
    v8f c = {};
    const float* arow = h + (size_t)(m0 + m) * HID;
    for (int k0 = 0; k0 < HID; k0 += 4) {
        int k = k0 + 2 * hh;
        v2f a; a.x = arow[k];               a.y = arow[k + 1];
        v2f b; b.x = W[k * C_OUT + m];      b.y = W[(k + 1) * C_OUT + m];
        c = __builtin_amdgcn_wmma_f32_16x16x4_f32(false, a, false, b,
                                                  (short)0, c, false, false);
    }
    float* drow = h2 + (size_t)(m0 + 8 * hh) * C_OUT + m;
#pragma unroll
    for (int i = 0; i < 8; ++i) drow[i * C_OUT] = c[i];
}

// ---------------- layer-1 edge scatter: agg1[col] += norm * h0[row] ----------------
__global__ void __launch_bounds__(256)
scatter1_kernel(const long long* __restrict__ ei, const float* __restrict__ ew,
                const float* __restrict__ dinv, const float* __restrict__ h0,
                float* agg) {
    __shared__ int   sr[4];
    __shared__ int   sc[4];
    __shared__ float sn[4];
    int tid = threadIdx.x;
    if (tid < 4) {
        int e = blockIdx.x * 4 + tid;
        int r = (int)ei[e];
        int c = (int)ei[N_EDGES + e];
        sr[tid] = r; sc[tid] = c;
        sn[tid] = dinv[r] * ew[e] * dinv[c];
    }
    __syncthreads();
    int j = tid >> 6;                             // edge within block
    int f = tid & 63;                             // feature -> coalesced
    float v = sn[j] * h0[(size_t)sr[j] * HID + f];
    atomicAdd(&agg[(size_t)sc[j] * HID + f], v);
}

// ---------------- layer-2 edge scatter (deleted edge weight -> 0) ----------------
__global__ void __launch_bounds__(256)
scatter2_kernel(const long long* __restrict__ ei, const float* __restrict__ ew,
                const float* __restrict__ dinv, const float* __restrict__ h2,
                float* agg, const int* __restrict__ del_idx) {
    __shared__ int   sr[16];
    __shared__ int   sc[16];
    __shared__ float sn[16];
    int tid = threadIdx.x;
    if (tid < 16) {
        int e = blockIdx.x * 16 + tid;
        int r = (int)ei[e];
        int c = (int)ei[N_EDGES + e];
        float w = (e == *del_idx) ? 0.0f : ew[e];
        sr[tid] = r; sc[tid] = c;
        sn[tid] = dinv[r] * w * dinv[c];
    }
    __syncthreads();
    int j = tid >> 4;
    int f = tid & 15;
    atomicAdd(&agg[(size_t)sc[j] * C_OUT + f],
              sn[j] * h2[(size_t)sr[j] * C_OUT + f]);
}

// ---------------- finalize: layer1 (relu, in-place h0 -> h) ----------------
__global__ void fin1_kernel(float* h0, const float* __restrict__ agg,
                            const float* __restrict__ dinv,
                            const float* __restrict__ b) {
    int gid = blockIdx.x * 256 + threadIdx.x;     // exactly N_NODES*HID threads
    int i = gid >> 6;
    int f = gid & 63;
    float s = dinv[i];
    float v = agg[gid] + s * s * h0[gid] + b[f];
    h0[gid] = fmaxf(v, 0.0f);
}

// ---------------- finalize: layer2 ----------------
__global__ void fin2_kernel(const float* __restrict__ h2, const float* __restrict__ agg,
                            const float* __restrict__ dinv,
                            const float* __restrict__ b, float* out) {
    int gid = blockIdx.x * 256 + threadIdx.x;     // exactly N_NODES*C_OUT threads
    int i = gid >> 4;
    int f = gid & 15;
    float s = dinv[i];
    out[gid] = agg[gid] + s * s * h2[gid] + b[f];
}

extern "C" void kernel_launch(void* const* d_in, const int* in_sizes, int n_in,
                              void* d_out, int out_size, void* d_ws, size_t ws_size,
                              hipStream_t stream) {
    const float*     x  = (const float*)d_in[0];
    const long long* ei = (const long long*)d_in[1];   // int64 edge_index [2, E]
    const float*     ew = (const float*)d_in[2];
    const float*     W1 = (const float*)d_in[3];
    const float*     b1 = (const float*)d_in[4];
    const float*     W2 = (const float*)d_in[5];
    const float*     b2 = (const float*)d_in[6];
    float* out = (float*)d_out;

    // workspace layout (floats); base is 256B aligned from hipMalloc
    float* wsf   = (float*)d_ws;
    float* deg   = wsf;                                   // 100000
    float* dinv1 = deg + N_NODES;                         // 100000
    float* dinv2 = dinv1 + N_NODES;                       // 100000
    unsigned long long* mp = (unsigned long long*)(dinv2 + N_NODES); // 8B (offset 1.2MB, aligned)
    int* del_idx = (int*)(mp + 1);                        // 4B
    float* h0    = wsf + 300004;                          // 6.4M (becomes h after fin1)
    float* agg1  = h0 + (size_t)N_NODES * HID;            // 6.4M
    float* h2    = agg1;                                  // reuse agg1 region after fin1
    float* agg2  = agg1 + (size_t)N_NODES * C_OUT;        // 1.6M, still inside agg1 region

    // degrees + argmin deleted edge
    init_deg_kernel <<<391,    256, 0, stream>>>(deg, mp);
    init_zero_kernel<<<25000,  256, 0, stream>>>(agg1, N_NODES * HID);
    edge_deg_kernel <<<6250,   256, 0, stream>>>(ei, ew, deg, mp);
    dinv_kernel     <<<391,    256, 0, stream>>>(deg, dinv1, dinv2);
    fixup_kernel    <<<1,      1,   0, stream>>>(ei, ew, deg, mp, dinv2, del_idx);

    // layer 1
    gemm1_wmma      <<<3125,   256, 0, stream>>>(x, W1, h0);
    scatter1_kernel <<<400000, 256, 0, stream>>>(ei, ew, dinv1, h0, agg1);
    fin1_kernel     <<<25000,  256, 0, stream>>>(h0, agg1, dinv1, b1);   // h0 -> h (relu)

    // layer 2 (agg1 region now dead -> reuse for h2/agg2)
    init_zero_kernel<<<6250,   256, 0, stream>>>(agg2, N_NODES * C_OUT);
    gemm2_wmma      <<<782,    256, 0, stream>>>(h0, W2, h2);
    scatter2_kernel <<<100000, 256, 0, stream>>>(ei, ew, dinv2, h2, agg2, del_idx);
    fin2_kernel     <<<6250,   256, 0, stream>>>(h2, agg2, dinv2, b2, out);
}